// ConvDecoder_35854386987036
// MI455X (gfx1250) — compile-verified
//
#include <hip/hip_runtime.h>
#include <math.h>

#define N_GRID 2048
#define NB 4
#define C_IN 16
#define C_OUT 32
#define N_CTX 256
#define N_OUT 1024

typedef __attribute__((ext_vector_type(2))) float v2f;
typedef __attribute__((ext_vector_type(8))) float v8f;

// ---------------------------------------------------------------------------
// Kernel 1: global min/max over x_context and x_target -> x_min, grid step
// ---------------------------------------------------------------------------
__global__ void minmax_kernel(const float* __restrict__ xc,
                              const float* __restrict__ xt,
                              float* __restrict__ ws) {
    __shared__ float smin[256];
    __shared__ float smax[256];
    const int t = threadIdx.x;
    float lo = 3.402823466e+38f, hi = -3.402823466e+38f;
    for (int i = t; i < NB * N_CTX; i += 256) {
        float v = xc[i]; lo = fminf(lo, v); hi = fmaxf(hi, v);
    }
    for (int i = t; i < NB * N_OUT; i += 256) {
        float v = xt[i]; lo = fminf(lo, v); hi = fmaxf(hi, v);
    }
    smin[t] = lo; smax[t] = hi;
    __syncthreads();
    for (int s = 128; s > 0; s >>= 1) {
        if (t < s) {
            smin[t] = fminf(smin[t], smin[t + s]);
            smax[t] = fmaxf(smax[t], smax[t + s]);
        }
        __syncthreads();
    }
    if (t == 0) {
        float xmin = smin[0] - 0.1f;
        float xmax = smax[0] + 0.1f;
        ws[0] = xmin;
        ws[1] = (xmax - xmin) / (float)(N_GRID - 1);
    }
}

// ---------------------------------------------------------------------------
// Kernel 2: z[b,o,c] = sum_i r[b,c,i] * exp(coef_c * (grid_i - x_t)^2)
// One thread per (o_local, c); one block per (b, 16-target tile).
// Transcendental-bound: independent fmaf(step,i,base) keeps v_exp_f32 fed.
// ---------------------------------------------------------------------------
__global__ void z_kernel(const float* __restrict__ r,
                         const float* __restrict__ xt,
                         const float* __restrict__ sigma,
                         const float* __restrict__ ws,
                         float* __restrict__ zout) {
    const int b     = blockIdx.x / (N_OUT / 16);
    const int otile = blockIdx.x % (N_OUT / 16);
    const int c  = threadIdx.x >> 4;   // 0..15  (channel)
    const int ol = threadIdx.x & 15;   // 0..15  (target within tile)
    const int o  = otile * 16 + ol;

    const float xmin = ws[0];
    const float step = ws[1];
    const float x    = xt[b * N_OUT + o];
    // coef = -0.5 / exp(sigma)^2 = -0.5 * exp(-2*sigma)
    const float coef = -0.5f * __expf(-2.0f * sigma[c]);
    const float base = xmin - x;

    const float* __restrict__ rr = r + (b * C_IN + c) * N_GRID;

    float acc = 0.0f;
#pragma unroll 8
    for (int i = 0; i < N_GRID; ++i) {
        float d = fmaf(step, (float)i, base);
        acc = fmaf(rr[i], __expf(d * d * coef), acc);
    }
    zout[(b * N_OUT + o) * C_IN + c] = acc;
}

// ---------------------------------------------------------------------------
// Kernel 3: out = z @ W + b via V_WMMA_F32_16X16X4_F32.
// One wave32 per 16-row tile of z (M=16); two N-tiles of 16 cover C_OUT=32;
// K=16 as 4 chained WMMAs. Bias preloaded into the f32 accumulator.
// A layout (16x4 f32): lane<16 -> M=lane, K={0,1}; lane>=16 -> M=lane-16, K={2,3}
// B layout (4x16 f32): lane<16 -> N=lane, K={0,1}; lane>=16 -> N=lane-16, K={2,3}
// C/D (16x16 f32):     VGPR v -> M = v + (lane<16 ? 0 : 8), N = lane%16
// ---------------------------------------------------------------------------
__global__ void gemm_wmma_kernel(const float* __restrict__ z,
                                 const float* __restrict__ W,
                                 const float* __restrict__ bias,
                                 float* __restrict__ out) {
    const int tile = blockIdx.x;        // rows tile*16 .. tile*16+15
    const int lane = threadIdx.x;       // 0..31, full wave active (EXEC all 1s)
    const int half = lane >> 4;         // 0 or 1
    const int lm   = lane & 15;
    const int row0 = tile * 16;

    // A operand pointer: row (row0+lm), starting K = half*2
    const float* __restrict__ zrow = z + (row0 + lm) * C_IN + half * 2;

#pragma unroll
    for (int ntile = 0; ntile < 2; ++ntile) {
        const int ncol = ntile * 16 + lm;
        const float bv = bias[ncol];
        v8f cacc;
#pragma unroll
        for (int v = 0; v < 8; ++v) cacc[v] = bv;

#pragma unroll
        for (int kk = 0; kk < C_IN; kk += 4) {
            v2f a;
            a.x = zrow[kk];
            a.y = zrow[kk + 1];
            v2f bm;
            const int kb = kk + half * 2;
            bm.x = W[kb * C_OUT + ncol];
            bm.y = W[(kb + 1) * C_OUT + ncol];
            cacc = __builtin_amdgcn_wmma_f32_16x16x4_f32(
                /*neg_a=*/false, a, /*neg_b=*/false, bm,
                /*c_mod=*/(short)0, cacc, /*reuse_a=*/false, /*reuse_b=*/false);
        }

#pragma unroll
        for (int v = 0; v < 8; ++v) {
            const int m = v + half * 8;
            out[(row0 + m) * C_OUT + ncol] = cacc[v];
        }
    }
}

// ---------------------------------------------------------------------------
// Launch
// ---------------------------------------------------------------------------
extern "C" void kernel_launch(void* const* d_in, const int* in_sizes, int n_in,
                              void* d_out, int out_size, void* d_ws, size_t ws_size,
                              hipStream_t stream) {
    const float* r     = (const float*)d_in[0];  // (B, C_IN, N_GRID)
    const float* xc    = (const float*)d_in[1];  // (B, N_CTX, 1)
    // d_in[2] = y_context: unused by the reference computation
    const float* xt    = (const float*)d_in[3];  // (B, N_OUT, 1)
    const float* sigma = (const float*)d_in[4];  // (C_IN,)
    const float* W     = (const float*)d_in[5];  // (C_IN, C_OUT)
    const float* bias  = (const float*)d_in[6];  // (C_OUT,)
    float* out = (float*)d_out;                  // (B, N_OUT, C_OUT)

    float* ws = (float*)d_ws;     // ws[0]=x_min, ws[1]=grid step
    float* zbuf = ws + 16;        // (B*N_OUT, C_IN) f32 = 256 KB

    minmax_kernel<<<1, 256, 0, stream>>>(xc, xt, ws);

    z_kernel<<<dim3(NB * (N_OUT / 16)), 256, 0, stream>>>(r, xt, sigma, ws, zbuf);

    gemm_wmma_kernel<<<dim3((NB * N_OUT) / 16), 32, 0, stream>>>(zbuf, W, bias, out);
}